// TransRLayer_27642409517751
// MI455X (gfx1250) — compile-verified
//
#include <hip/hip_runtime.h>
#include <hip/hip_bf16.h>

#define EMB 64
#define NREL 100
#define RELPAD 128
#define MAX_TILES 4224   // 100 relations + 65536/16 full tiles, padded

typedef float v2f __attribute__((ext_vector_type(2)));
typedef float v8f __attribute__((ext_vector_type(8)));
typedef unsigned int u32x4 __attribute__((ext_vector_type(4)));
typedef int i32x4 __attribute__((ext_vector_type(4)));
typedef int i32x8 __attribute__((ext_vector_type(8)));

// ---------------- pass 0: zero counters ----------------
__global__ void zero_k(int* ws) {
    int i = threadIdx.x;           // one block of 512
    ws[i] = 0;                     // counts[128], offs[128], cursor[128], ntiles+pad
}

// ---------------- pass 1: histogram ----------------
__global__ void hist_k(const int* __restrict__ rel, int n, int* __restrict__ counts) {
    int i = blockIdx.x * blockDim.x + threadIdx.x;
    if (i < n) atomicAdd(&counts[rel[i]], 1);
}

// ---------------- pass 2: prefix + tile plan ----------------
__global__ void plan_k(int* __restrict__ counts, int* __restrict__ offs,
                       int* __restrict__ cursor, int* __restrict__ ntiles,
                       int* __restrict__ tile_rel, int* __restrict__ tile_row,
                       int* __restrict__ tile_cnt) {
    __shared__ int toff[NREL + 1];
    if (threadIdx.x == 0) {
        int acc = 0, tacc = 0;
        for (int r = 0; r < NREL; ++r) {
            offs[r] = acc; cursor[r] = acc; toff[r] = tacc;
            acc  += counts[r];
            tacc += (counts[r] + 15) >> 4;
        }
        offs[NREL] = acc; toff[NREL] = tacc; *ntiles = tacc;
    }
    __syncthreads();
    for (int r = threadIdx.x; r < NREL; r += blockDim.x) {
        int c = counts[r], base = offs[r], tb = toff[r];
        int nt = (c + 15) >> 4;
        for (int i = 0; i < nt; ++i) {
            tile_rel[tb + i] = r;
            tile_row[tb + i] = base + i * 16;
            tile_cnt[tb + i] = min(16, c - i * 16);
        }
    }
}

// ---------------- pass 3: scatter permutation ----------------
__global__ void scatter_k(const int* __restrict__ rel, int n,
                          int* __restrict__ cursor, int* __restrict__ perm) {
    int i = blockIdx.x * blockDim.x + threadIdx.x;
    if (i < n) {
        int p = atomicAdd(&cursor[rel[i]], 1);
        perm[p] = i;
    }
}

// ---------------- pass 4: grouped GEMM via WMMA f32 16x16x4 + TDM ----------------
// Block = 128 threads = 4 waves. One block per 16-row tile sharing a relation.
// Wave w computes output columns [16w, 16w+16).
__global__ __launch_bounds__(128) void gemm_k(
    const int* __restrict__ head, const int* __restrict__ tail,
    const float* __restrict__ entity, const float* __restrict__ relemb,
    const float* __restrict__ proj,
    const int* __restrict__ perm, const int* __restrict__ tile_rel,
    const int* __restrict__ tile_row, const int* __restrict__ tile_cnt,
    const int* __restrict__ ntiles, float* __restrict__ out)
{
    __shared__ float dvec[16][68];   // (h - t) rows, padded: bank = (4m + k) % 64
    __shared__ float pmat[64][72];   // proj; row stride 72 = 64 data + 8 pad DWORDs (TDM pad)
    __shared__ float remb_s[EMB];
    __shared__ float rowsum[16];
    __shared__ int   hb[16], tb16[16];

    int t = blockIdx.x;
    if (t >= *ntiles) return;
    int r    = tile_rel[t];
    int row0 = tile_row[t];
    int cnt  = tile_cnt[t];
    int tid  = threadIdx.x;

    const float* pbase = proj + (size_t)r * (EMB * EMB);

#if __has_builtin(__builtin_amdgcn_tensor_load_to_lds)
    // Tensor Data Mover: DMA the 64x64 f32 proj tile into pmat.
    // Padding: after every 64 DWORDs insert 8 DWORDs -> LDS row stride 72 floats.
    {
        unsigned lds_off = (unsigned)(uintptr_t)(void*)&pmat[0][0];
        unsigned long long ga = (unsigned long long)(uintptr_t)pbase;
        u32x4 g0;
        g0.x = 1u;                                                  // count=1 (valid user D#)
        g0.y = lds_off;                                             // lds_addr (bytes)
        g0.z = (unsigned)ga;                                        // global_addr[31:0]
        g0.w = (unsigned)((ga >> 32) & 0x1FFFFFFull) | (2u << 30);  // addr[56:32] | type=2
        i32x8 g1;
        g1[0] = (2 << 16) | (1 << 20) | (5 << 22) | (7 << 25);      // 4B elems, pad_en, 64DW/8DW
        g1[1] = (int)(64u << 16);                                   // tensor_dim0 = 64
        g1[2] = (int)(64u << 16);                                   // tensor_dim1 = 64
        g1[3] = (int)(64u << 16);                                   // tile_dim0 = 64
        g1[4] = 64;                                                 // tile_dim1 = 64, tile_dim2 = 0
        g1[5] = 64;                                                 // tensor_dim0_stride = 64
        g1[6] = 0;
        g1[7] = 0;
        i32x4 g2 = {1, 0, 0, 0};                                    // tensor_dim2=1; dims unused
        i32x4 g3 = {0, 0, 0, 0};
        i32x8 g4 = {0, 0, 0, 0, 0, 0, 0, 0};                        // extra group (clang-23 form)
        if (tid < 32) {
            __builtin_amdgcn_tensor_load_to_lds(g0, g1, g2, g3, g4, 0);
            __builtin_amdgcn_s_wait_tensorcnt((short)0);
        }
    }
#endif
    __builtin_prefetch(pbase, 0, 1);   // global_prefetch_b8

    if (tid < 16) {
        int b = (tid < cnt) ? perm[row0 + tid] : -1;
        hb[tid]   = (b >= 0) ? head[b] * EMB : -1;
        tb16[tid] = (b >= 0) ? tail[b] * EMB : -1;
        rowsum[tid] = 0.0f;
    }
    if (tid < EMB) remb_s[tid] = relemb[r * EMB + tid];
    __syncthreads();

    // stage A = h - t  (16 x 64), 8 floats/thread
    for (int idx = tid; idx < 16 * EMB; idx += 128) {
        int m = idx >> 6, k = idx & 63;
        float v = 0.0f;
        int hbm = hb[m];
        if (hbm >= 0) v = entity[hbm + k] - entity[tb16[m] + k];
        dvec[m][k] = v;
    }
#if !__has_builtin(__builtin_amdgcn_tensor_load_to_lds)
    // fallback staging: vector loads + ds stores
    for (int idx = tid; idx < EMB * EMB; idx += 128) {
        pmat[idx >> 6][idx & 63] = pbase[idx];
    }
#endif
    __syncthreads();

    int wave = tid >> 5;
    int lane = tid & 31;
    int half = lane >> 4;        // 0: K, K+1   1: K+2, K+3 (A/B layout)
    int l16  = lane & 15;
    int col  = wave * 16 + l16;  // N index (B and C layout: N = lane % 16)

    v8f acc = {0.f, 0.f, 0.f, 0.f, 0.f, 0.f, 0.f, 0.f};
    #pragma unroll
    for (int k0 = 0; k0 < EMB; k0 += 4) {
        int ka = k0 + 2 * half;
        v2f a, b;
        a.x = dvec[l16][ka];
        a.y = dvec[l16][ka + 1];
        b.x = pmat[ka][col];
        b.y = pmat[ka + 1][col];
        acc = __builtin_amdgcn_wmma_f32_16x16x4_f32(
            /*neg_a=*/false, a, /*neg_b=*/false, b,
            /*c_mod=*/(short)0, acc, /*reuse_a=*/false, /*reuse_b=*/false);
    }

    // acc[i] holds D[row = i + 8*half][col]; add r_emb, square, reduce rows
    float re = remb_s[col];
    #pragma unroll
    for (int i = 0; i < 8; ++i) {
        float v = acc[i] + re;
        float s = v * v;
        s += __shfl_xor(s, 1, 32);
        s += __shfl_xor(s, 2, 32);
        s += __shfl_xor(s, 4, 32);
        s += __shfl_xor(s, 8, 32);   // all lanes in each 16-group hold group sum
        if (l16 == 0) atomicAdd(&rowsum[i + 8 * half], s);
    }
    __syncthreads();
    if (tid < cnt) out[perm[row0 + tid]] = rowsum[tid];
}

// ---------------- fallback (ws too small): pure VALU ----------------
__global__ __launch_bounds__(64) void naive_k(
    const int* __restrict__ head, const int* __restrict__ rel,
    const int* __restrict__ tail, const float* __restrict__ entity,
    const float* __restrict__ relemb, const float* __restrict__ proj,
    float* __restrict__ out)
{
    __shared__ float red[64];
    int b = blockIdx.x, j = threadIdx.x;
    int rr = rel[b];
    const float* ph = entity + head[b] * EMB;
    const float* pt = entity + tail[b] * EMB;
    const float* pp = proj + (size_t)rr * (EMB * EMB);
    float acc = relemb[rr * EMB + j];
    #pragma unroll 8
    for (int e = 0; e < EMB; ++e) acc += (ph[e] - pt[e]) * pp[e * EMB + j];
    red[j] = acc * acc;
    __syncthreads();
    if (j < 32) red[j] += red[j + 32];
    __syncthreads();
    if (j == 0) {
        float s = 0.f;
        for (int k = 0; k < 32; ++k) s += red[k];
        out[b] = s;
    }
}

extern "C" void kernel_launch(void* const* d_in, const int* in_sizes, int n_in,
                              void* d_out, int out_size, void* d_ws, size_t ws_size,
                              hipStream_t stream) {
    const int*   head   = (const int*)d_in[0];
    const int*   rel    = (const int*)d_in[1];
    const int*   tail   = (const int*)d_in[2];
    const float* entity = (const float*)d_in[3];
    const float* relemb = (const float*)d_in[4];
    const float* proj   = (const float*)d_in[5];
    float*       out    = (float*)d_out;

    int batch = in_sizes[0];
    size_t needed = (size_t)(512 + batch + 3 * MAX_TILES) * sizeof(int);

    if (ws_size < needed) {
        naive_k<<<batch, 64, 0, stream>>>(head, rel, tail, entity, relemb, proj, out);
        return;
    }

    int* ws       = (int*)d_ws;
    int* counts   = ws;
    int* offs     = ws + 128;
    int* cursor   = ws + 256;
    int* ntiles   = ws + 384;
    int* perm     = ws + 512;
    int* tile_rel = perm + batch;
    int* tile_row = tile_rel + MAX_TILES;
    int* tile_cnt = tile_row + MAX_TILES;

    int grid = (batch + 255) / 256;
    zero_k   <<<1, 512, 0, stream>>>(ws);
    hist_k   <<<grid, 256, 0, stream>>>(rel, batch, counts);
    plan_k   <<<1, 128, 0, stream>>>(counts, offs, cursor, ntiles,
                                     tile_rel, tile_row, tile_cnt);
    scatter_k<<<grid, 256, 0, stream>>>(rel, batch, cursor, perm);
    gemm_k   <<<MAX_TILES, 128, 0, stream>>>(head, tail, entity, relemb, proj,
                                             perm, tile_rel, tile_row, tile_cnt,
                                             ntiles, out);
}